// AgreementRouting_57913339019364
// MI455X (gfx1250) — compile-verified
//
#include <hip/hip_runtime.h>
#include <math.h>

#define B_SZ     512
#define IN_CAPS  1152
#define OUT_CAPS 10
#define OUT_DIM  16

typedef __attribute__((ext_vector_type(2))) float v2f;
typedef __attribute__((ext_vector_type(8))) float v8f;

// ---------------------------------------------------------------------------
// Kernel 1: c[i,j] = softmax_j(b[i,j])   (b is (1152,10); batch-invariant)
// ---------------------------------------------------------------------------
__global__ void softmax_c_kernel(const float* __restrict__ b,
                                 float* __restrict__ c) {
    int i = blockIdx.x * blockDim.x + threadIdx.x;
    if (i >= IN_CAPS) return;
    float x[OUT_CAPS];
    float m = b[i * OUT_CAPS + 0];
    x[0] = m;
    #pragma unroll
    for (int j = 1; j < OUT_CAPS; ++j) {
        x[j] = b[i * OUT_CAPS + j];
        m = fmaxf(m, x[j]);
    }
    float s = 0.0f;
    #pragma unroll
    for (int j = 0; j < OUT_CAPS; ++j) {
        x[j] = expf(x[j] - m);
        s += x[j];
    }
    float inv = 1.0f / s;
    #pragma unroll
    for (int j = 0; j < OUT_CAPS; ++j)
        c[i * OUT_CAPS + j] = x[j] * inv;
}

// ---------------------------------------------------------------------------
// Kernel 2: one wave per (batch b, out-capsule j).
// s[b,j,d] = sum_i c[i,j] * u[b,i,j,d] via V_WMMA_F32_16X16X4_F32 with a
// broadcast A operand; then squash across d with lane shuffles.
// u layout: ((b*1152 + i)*10 + j)*16 + d
// ---------------------------------------------------------------------------
__global__ void __launch_bounds__(320)
routing_wmma_kernel(const float* __restrict__ u,
                    const float* __restrict__ c,
                    float* __restrict__ out) {
    __shared__ float c_lds[IN_CAPS * OUT_CAPS];
    for (int t = threadIdx.x; t < IN_CAPS * OUT_CAPS; t += 320)
        c_lds[t] = c[t];
    __syncthreads();

    const int b    = blockIdx.x;
    const int j    = threadIdx.x >> 5;   // wave id -> output capsule (0..9)
    const int lane = threadIdx.x & 31;
    const int d    = lane & 15;          // N index (output dim)
    const int hi   = lane >> 4;          // 0: lanes 0-15, 1: lanes 16-31

    // Half-wave K striping per ISA 16x4 f32 A layout:
    // VGPR0 holds K={0,2}, VGPR1 holds K={1,3}
    const int k0 = hi ? 2 : 0;
    const int k1 = hi ? 3 : 1;

    const float* ub = u + (size_t)b * (IN_CAPS * OUT_CAPS * OUT_DIM)
                        + j * OUT_DIM + d;
    const float* cj = c_lds + j;

    v8f acc0 = {};
    v8f acc1 = {};

    for (int i0 = 0; i0 < IN_CAPS; i0 += 8) {
        v2f a0, b0, a1, b1;
        // K-chunk i0..i0+3
        a0.x = cj[(i0 + k0) * OUT_CAPS];
        a0.y = cj[(i0 + k1) * OUT_CAPS];
        b0.x = ub[(size_t)(i0 + k0) * (OUT_CAPS * OUT_DIM)];
        b0.y = ub[(size_t)(i0 + k1) * (OUT_CAPS * OUT_DIM)];
        // K-chunk i0+4..i0+7
        a1.x = cj[(i0 + 4 + k0) * OUT_CAPS];
        a1.y = cj[(i0 + 4 + k1) * OUT_CAPS];
        b1.x = ub[(size_t)(i0 + 4 + k0) * (OUT_CAPS * OUT_DIM)];
        b1.y = ub[(size_t)(i0 + 4 + k1) * (OUT_CAPS * OUT_DIM)];

        // D = A*B + C ; rows of D are identical (A rows are identical)
        acc0 = __builtin_amdgcn_wmma_f32_16x16x4_f32(
            false, a0, false, b0, (short)0, acc0, false, false);
        acc1 = __builtin_amdgcn_wmma_f32_16x16x4_f32(
            false, a1, false, b1, (short)0, acc1, false, false);
    }

    // Row 0 of D: lanes 0-15 hold s[b,j,d] in acc[0]
    float sv = acc0[0] + acc1[0];

    // squash: l2 = sum_d s^2 ; scale = l2/(1+l2)/sqrt(l2)
    float sq = sv * sv;
    sq += __shfl_xor(sq, 1);
    sq += __shfl_xor(sq, 2);
    sq += __shfl_xor(sq, 4);
    sq += __shfl_xor(sq, 8);

    float l2    = sq;
    float scale = l2 / (1.0f + l2) / sqrtf(l2);

    if (lane < 16)
        out[(b * OUT_CAPS + j) * OUT_DIM + lane] = sv * scale;
}

// ---------------------------------------------------------------------------
extern "C" void kernel_launch(void* const* d_in, const int* in_sizes, int n_in,
                              void* d_out, int out_size, void* d_ws, size_t ws_size,
                              hipStream_t stream) {
    const float* u    = (const float*)d_in[0];   // (512,1152,10,16) f32
    const float* bvec = (const float*)d_in[1];   // (1152,10) f32
    // d_in[2] = n_iterations: does not affect the returned values (the loop
    // only updates b_batch, which is dead for the output) -> ignored.

    float* c = (float*)d_ws;                     // (1152,10) routing coefficients

    softmax_c_kernel<<<(IN_CAPS + 255) / 256, 256, 0, stream>>>(bvec, c);
    routing_wmma_kernel<<<B_SZ, 320, 0, stream>>>(u, c, (float*)d_out);
}